// QuantumNeuralLayer_54640573940369
// MI455X (gfx1250) — compile-verified
//
#include <hip/hip_runtime.h>
#include <math.h>

typedef __attribute__((ext_vector_type(2))) float v2f;
typedef __attribute__((ext_vector_type(8))) float v8f;

#define INV_SQRT2 0.70710678118654752440f
#define O_DIM 64
#define K_DIM 256
#define LDS_STRIDE 260   // 260 % 64 == 4 -> conflict-free b64 B-fragment reads
#define ROWS_PER_WAVE 64
#define ROWS_PER_BLOCK 512

// ---------------------------------------------------------------------------
// Kernel 1: the "quantum" term is batch-independent (state always starts at
// |0>), so fold it into the bias once: bias_out[o] = b[o] + 0.1 * q[o].
// One block of 64 threads; thread o owns output channel o.
// ---------------------------------------------------------------------------
__global__ void quantum_bias_kernel(const float* __restrict__ b,
                                    const float* __restrict__ phase,  // [D,O]
                                    const float* __restrict__ ent,    // [O,O]
                                    float* __restrict__ bias_out,     // [O]
                                    int D) {
    __shared__ float sa[O_DIM];
    __shared__ float sc[O_DIM];
    const int o = threadIdx.x;
    float a = 1.0f, c = 0.0f;                    // state |0> per channel
    for (int d = 0; d < D; ++d) {
        const float ph = phase[d * O_DIM + o];
        const float ha = (a + c) * INV_SQRT2 * sinf(ph);
        const float hc = (a - c) * INV_SQRT2 * cosf(ph);
        sa[o] = ha; sc[o] = hc;
        __syncthreads();
        float na = 0.0f, nc = 0.0f;
        for (int i = 0; i < O_DIM; ++i) {
            const float e = ent[i * O_DIM + o]; // s_new[o] = sum_i s[i]*E[i,o]
            na += sa[i] * e;
            nc += sc[i] * e;
        }
        __syncthreads();
        a = na; c = nc;
    }
    bias_out[o] = b[o] + 0.1f * (a * a + c * c);
}

// ---------------------------------------------------------------------------
// Kernel 2: FP32 WMMA GEMM  out[B,64] = x[B,256] @ W^T + bias
// Wave tile: 64(M) x 64(N) = 4x4 grid of 16x16x4 f32 WMMA accumulators.
// W (64KB) staged in LDS with padded stride; x streams from HBM.
// ---------------------------------------------------------------------------
__global__ void __launch_bounds__(256, 1)
gemm_bias_kernel(const float* __restrict__ x,
                 const float* __restrict__ W,      // [64,256] row-major
                 const float* __restrict__ biasc,  // [64] combined bias
                 float* __restrict__ out,          // [B,64]
                 int Btot) {
    __shared__ float ldsW[O_DIM * LDS_STRIDE];    // ~65 KB of 320 KB/WGP

    // Cooperative copy of W into LDS (pad each 256-float row to 260 floats).
    {
        const float2* W2 = reinterpret_cast<const float2*>(W);
        const int npairs = O_DIM * K_DIM / 2;
        for (int e = threadIdx.x; e < npairs; e += blockDim.x) {
            float2 v = W2[e];
            int idx = e * 2;
            int row = idx >> 8;      // /256
            int col = idx & 255;
            ldsW[row * LDS_STRIDE + col]     = v.x;
            ldsW[row * LDS_STRIDE + col + 1] = v.y;
        }
    }
    __syncthreads();

    const int lane = threadIdx.x & 31;
    const int wave = threadIdx.x >> 5;
    const int grp  = lane >> 4;      // half-wave group (selects K pair)
    const int nl   = lane & 15;      // row index (A) / column index (B,D)
    const long rowBase = (long)blockIdx.x * ROWS_PER_BLOCK + (long)wave * ROWS_PER_WAVE;

    if (rowBase >= Btot) return;

    if (rowBase + ROWS_PER_WAVE > Btot) {
        // Scalar tail for a partial tile (not hit when B % 512 == 0).
        for (long r = rowBase; r < Btot; ++r) {
            const float* xr = x + r * K_DIM;
            float s0 = 0.0f, s1 = 0.0f;
            for (int k = 0; k < K_DIM; ++k) {
                const float xv = xr[k];
                s0 += xv * ldsW[lane * LDS_STRIDE + k];
                s1 += xv * ldsW[(lane + 32) * LDS_STRIDE + k];
            }
            out[r * O_DIM + lane]      = s0 + biasc[lane];
            out[r * O_DIM + lane + 32] = s1 + biasc[lane + 32];
        }
        return;
    }

    v8f acc[4][4];
    const v8f vzero = {0.f, 0.f, 0.f, 0.f, 0.f, 0.f, 0.f, 0.f};
#pragma unroll
    for (int m = 0; m < 4; ++m)
#pragma unroll
        for (int n = 0; n < 4; ++n)
            acc[m][n] = vzero;

    const float* xw = x + rowBase * K_DIM;

#pragma unroll 2
    for (int kb = 0; kb < K_DIM; kb += 4) {
        const int kA = kb + 2 * grp;   // each lane holds K = {kA, kA+1}

        // A fragments: 16x4 f32, lane -> row (lane&15), 2 consecutive k's.
        v2f afrag[4];
#pragma unroll
        for (int m = 0; m < 4; ++m)
            afrag[m] = *(const v2f*)(xw + (long)(m * 16 + nl) * K_DIM + kA);

        // B fragments: 4x16 f32 from LDS, lane -> col (lane&15), same k pair.
        v2f bfrag[4];
#pragma unroll
        for (int n = 0; n < 4; ++n)
            bfrag[n] = *(const v2f*)(ldsW + (n * 16 + nl) * LDS_STRIDE + kA);

#pragma unroll
        for (int m = 0; m < 4; ++m)
#pragma unroll
            for (int n = 0; n < 4; ++n)
                acc[m][n] = __builtin_amdgcn_wmma_f32_16x16x4_f32(
                    /*neg_a=*/false, afrag[m],
                    /*neg_b=*/false, bfrag[n],
                    /*c_mod=*/(short)0, acc[m][n],
                    /*reuse_a=*/false, /*reuse_b=*/false);
    }

    // Store: D layout -> lane holds N = (lane&15), VGPR j holds M = j + 8*grp.
#pragma unroll
    for (int n = 0; n < 4; ++n) {
        const int col = n * 16 + nl;
        const float bias = biasc[col];
#pragma unroll
        for (int m = 0; m < 4; ++m) {
            const v8f r = acc[m][n];
            const long row0 = rowBase + m * 16 + grp * 8;
#pragma unroll
            for (int j = 0; j < 8; ++j)
                out[(row0 + j) * O_DIM + col] = r[j] + bias;
        }
    }
}

// ---------------------------------------------------------------------------
extern "C" void kernel_launch(void* const* d_in, const int* in_sizes, int n_in,
                              void* d_out, int out_size, void* d_ws, size_t ws_size,
                              hipStream_t stream) {
    const float* x   = (const float*)d_in[0];  // [B,256]
    const float* W   = (const float*)d_in[1];  // [64,256]
    const float* b   = (const float*)d_in[2];  // [64]
    const float* ps  = (const float*)d_in[3];  // [D,64]
    const float* ent = (const float*)d_in[4];  // [64,64]
    float* out = (float*)d_out;
    float* ws  = (float*)d_ws;                 // 64-float combined bias

    const int Btot = in_sizes[0] / K_DIM;
    const int D    = in_sizes[3] / O_DIM;

    quantum_bias_kernel<<<1, O_DIM, 0, stream>>>(b, ps, ent, ws, D);

    const int blocks = (Btot + ROWS_PER_BLOCK - 1) / ROWS_PER_BLOCK;
    gemm_bias_kernel<<<blocks, 256, 0, stream>>>(x, W, ws, out, Btot);
}